// SimplePredictionWrapper_16157666967690
// MI455X (gfx1250) — compile-verified
//
#include <hip/hip_runtime.h>
#include <stdint.h>

#define VOCAB     128000
#define NTHREADS  640                   // 20 wave32 waves -> 5 per SIMD32
#define VEC       4
#define TILE      (NTHREADS * VEC)      // 2560 elements per iteration
#define NITER     (VOCAB / TILE)        // 50 exactly
#define TOPK      50
#define TOPP      0.9f
#define NEGV      (-1000000000.0f)
#define NBINS     4096
#define CHUNK     16
#define NCHUNK    (NBINS / CHUNK)       // 256
#define CANDMAX   512
#define PADV      (-3.0e38f)

typedef float f32x4 __attribute__((ext_vector_type(4)));   // native clang vector (builtin-friendly)

// Order-preserving map: float -> uint32 so that x >= y  <=>  key(x) >= key(y)
__device__ __forceinline__ uint32_t order_key(float x) {
  uint32_t u = __float_as_uint(x);
  return (u & 0x80000000u) ? ~u : (u | 0x80000000u);
}

__global__ __launch_bounds__(NTHREADS, 1)
void topk_topp_filter(const float* __restrict__ logits, float* __restrict__ out) {
  __shared__ int      s_hist[NBINS];
  __shared__ int      s_chunk[NCHUNK];
  __shared__ float    s_cand[CANDMAX];
  __shared__ f32x4    s_stage[2][NTHREADS];   // async-load staging, double buffered
  __shared__ int      s_ncand;
  __shared__ uint32_t s_binT;
  __shared__ float    s_thresh;

  const int tid = threadIdx.x;
  const int row = blockIdx.x;
  const float*  rp  = logits + (size_t)row * VOCAB;
  const f32x4*  rp4 = (const f32x4*)rp;
  f32x4*        op4 = (f32x4*)(out + (size_t)row * VOCAB);

  for (int i = tid; i < NBINS; i += NTHREADS) s_hist[i] = 0;
  if (tid == 0) s_ncand = 0;
  __syncthreads();

  // ---------------- Pass A: 12-bit radix histogram ----------------
  // Stream HBM -> LDS via CDNA5 async global->LDS loads (ASYNCcnt-tracked),
  // double buffered: next tile in flight while histogramming the current one.
  uint32_t lds0 = (uint32_t)(uintptr_t)&s_stage[0][tid];  // low 32 bits == LDS byte offset
  uint32_t lds1 = (uint32_t)(uintptr_t)&s_stage[1][tid];
  {
    const float* g0 = rp + (size_t)tid * VEC;
    asm volatile("global_load_async_to_lds_b128 %0, %1, off"
                 :: "v"(lds0), "v"(g0) : "memory");
  }
  for (int it = 0; it < NITER; ++it) {
    if (it + 1 < NITER) {
      const float* gn = rp + (size_t)(it + 1) * TILE + (size_t)tid * VEC;
      uint32_t ldsn = ((it + 1) & 1) ? lds1 : lds0;
      asm volatile("global_load_async_to_lds_b128 %0, %1, off"
                   :: "v"(ldsn), "v"(gn) : "memory");
      asm volatile("s_wait_asynccnt 0x1" ::: "memory");   // oldest tile landed
    } else {
      asm volatile("s_wait_asynccnt 0x0" ::: "memory");   // drain
    }
    f32x4 v = s_stage[it & 1][tid];
    atomicAdd(&s_hist[order_key(v.x) >> 20], 1);
    atomicAdd(&s_hist[order_key(v.y) >> 20], 1);
    atomicAdd(&s_hist[order_key(v.z) >> 20], 1);
    atomicAdd(&s_hist[order_key(v.w) >> 20], 1);
  }
  __syncthreads();

  // ---------------- Two-level scan to find the 50th-largest value's bin ----------------
  // Level 1 (parallel): 256 threads each sum a 16-bin chunk.
  if (tid < NCHUNK) {
    int s = 0;
#pragma unroll
    for (int b = 0; b < CHUNK; ++b) s += s_hist[tid * CHUNK + b];
    s_chunk[tid] = s;
  }
  __syncthreads();
  // Level 2 (serial, but only ~272 steps instead of 4096): locate chunk, then bin.
  if (tid == 0) {
    int cum = 0;
    int c = NCHUNK - 1;
    for (; c > 0; --c) {
      int v = s_chunk[c];
      if (cum + v >= TOPK) break;
      cum += v;
    }
    uint32_t bin = (uint32_t)(c * CHUNK);
    for (int b = c * CHUNK + CHUNK - 1; b >= c * CHUNK; --b) {
      int v = s_hist[b];
      if (cum + v >= TOPK) { bin = (uint32_t)b; break; }
      cum += v;
    }
    s_binT = bin;
  }
  __syncthreads();
  const uint32_t binT = s_binT;

  // ---------------- Pass B: gather candidates (key-bin >= binT); L2-resident ----------------
  for (int it = 0; it < NITER; ++it) {
    f32x4 v = rp4[it * NTHREADS + tid];
    float xs[4] = {v.x, v.y, v.z, v.w};
#pragma unroll
    for (int e = 0; e < 4; ++e) {
      if ((order_key(xs[e]) >> 20) >= binT) {
        int pos = atomicAdd(&s_ncand, 1);
        if (pos < CANDMAX) s_cand[pos] = xs[e];
      }
    }
  }
  __syncthreads();
  int ncand = s_ncand; if (ncand > CANDMAX) ncand = CANDMAX;
  for (int i = tid; i < CANDMAX; i += NTHREADS)
    if (i >= ncand) s_cand[i] = PADV;
  __syncthreads();

  // ---------------- Bitonic sort (descending), 512 slots ----------------
  for (int k = 2; k <= CANDMAX; k <<= 1) {
    for (int j = k >> 1; j > 0; j >>= 1) {
      for (int i = tid; i < CANDMAX; i += NTHREADS) {
        int l = i ^ j;
        if (l > i) {
          float a = s_cand[i], b = s_cand[l];
          bool desc = ((i & k) == 0);
          bool sw = desc ? (a < b) : (a > b);
          if (sw) { s_cand[i] = b; s_cand[l] = a; }
        }
      }
      __syncthreads();
    }
  }

  // ---------------- Top-p over the top-K prefix -> single scalar threshold ----------------
  if (tid == 0) {
    float mx  = s_cand[0];
    float sum = 0.f;
    for (int i = 0; i < TOPK; ++i) sum += __expf(s_cand[i] - mx);
    float limit = TOPP * sum;
    float cum = __expf(s_cand[0] - mx);   // position 0 always kept
    int m = 1;
    for (int i = 1; i < TOPK; ++i) {      // keep i iff cum_{i-1} <= p
      if (cum > limit) break;
      m = i + 1;
      cum += __expf(s_cand[i] - mx);
    }
    s_thresh = s_cand[m - 1];             // value of last kept token
  }
  __syncthreads();
  const float th = s_thresh;

  // ---------------- Pass C: threshold and write ----------------
  // Reads hit L2 (input is resident); stores are non-temporal so the 131 MB of
  // write-once output does not evict the L2-resident input of other rows.
  for (int it = 0; it < NITER; ++it) {
    f32x4 v = rp4[it * NTHREADS + tid];
    f32x4 o;
    o.x = (v.x >= th) ? v.x : NEGV;
    o.y = (v.y >= th) ? v.y : NEGV;
    o.z = (v.z >= th) ? v.z : NEGV;
    o.w = (v.w >= th) ? v.w : NEGV;
    __builtin_nontemporal_store(o, &op4[it * NTHREADS + tid]);
  }
}

extern "C" void kernel_launch(void* const* d_in, const int* in_sizes, int n_in,
                              void* d_out, int out_size, void* d_ws, size_t ws_size,
                              hipStream_t stream) {
  (void)n_in; (void)out_size; (void)d_ws; (void)ws_size;
  const float* logits = (const float*)d_in[0];
  float* out = (float*)d_out;
  int rows = in_sizes[0] / VOCAB;   // 32*8 = 256 rows
  topk_topp_filter<<<dim3(rows), dim3(NTHREADS), 0, stream>>>(logits, out);
}